// NgramMinPooling_10033043603712
// MI455X (gfx1250) — compile-verified
//
#include <hip/hip_runtime.h>
#include <stdint.h>

// Problem geometry (fixed by the reference).
#define BB 32
#define SS 2048
#define HH 512
#define ROWS_TOTAL (BB * SS)          // 65536 rows
#define CHUNK 16                      // rows computed per block
#define HALO 2                        // N_GRAM - 1
#define LDS_ROWS (CHUNK + HALO)       // 18 rows staged (36 KB)
#define F4_PER_ROW (HH / 4)           // 128 float4 per row
#define THREADS 256                   // 8 waves (wave32)

typedef __attribute__((ext_vector_type(4))) unsigned int v4u;
typedef __attribute__((ext_vector_type(8))) int          v8i;
typedef __attribute__((ext_vector_type(4))) int          v4i;
typedef __attribute__((ext_vector_type(4))) float        v4f;  // clang vector,
// legal for __builtin_nontemporal_store (HIP's float4 struct is not)

#if __has_include(<hip/amd_detail/amd_gfx1250_TDM.h>)
#define TDM_6ARG 1
#endif

#if defined(__has_builtin)
#if !__has_builtin(__builtin_amdgcn_tensor_load_to_lds)
#define NO_TDM 1
#endif
#else
#define NO_TDM 1
#endif

// ---------------------------------------------------------------- helpers ---
__device__ __forceinline__ float sigmoid_f(float v) {
  // numerically stable, matches jax.nn.sigmoid in fp32
  if (v >= 0.0f) {
    float e = expf(-v);
    return 1.0f / (1.0f + e);
  }
  float e = expf(v);
  return e / (1.0f + e);
}

// ------------------------------------------------------------ fused kernel --
// Placed FIRST in the file so the driver's asm snippet shows the TDM codegen.
__global__ void __launch_bounds__(THREADS)
ngram_min_fused(const float* __restrict__ x,
                const unsigned* __restrict__ bm,
                float* __restrict__ out) {
  __shared__ float sm[LDS_ROWS * HH];  // 36 KB

  const int r0 = blockIdx.x * CHUNK;
  const int gbase = (r0 >= HALO) ? (r0 - HALO) : 0;  // first staged row

#if !defined(NO_TDM)
  // ---- Stage 18 contiguous rows (gbase .. gbase+17) into LDS via the TDM.
  // One 1-D descriptor: data_size=4B, tile_dim0 = tensor_dim0 = 9216 elems.
  if (threadIdx.x < 32) {  // wave 0 only issues the DMA (EXEC ignored by TDM)
    unsigned long long ga =
        (unsigned long long)(uintptr_t)(const void*)(x + (size_t)gbase * HH);
    unsigned ldsa = (unsigned)(uintptr_t)(void*)&sm[0];  // LDS byte address
    const unsigned nelem = (unsigned)(LDS_ROWS * HH);    // 9216

    v4u g0;  // D# group 0: count=1 | lds_addr | global_addr | type=2
    g0[0] = 1u;                                   // count=1, is_restore=0
    g0[1] = ldsa;                                 // lds_addr[31:0]
    g0[2] = (unsigned)(ga & 0xFFFFFFFFull);       // global_addr[31:0]
    g0[3] = (unsigned)((ga >> 32) & 0x01FFFFFFull) | 0x80000000u; // [56:32]|type=2

    v8i g1;  // D# group 1
    g1[0] = 0x00020000;                     // wg_mask=0, data_size=2 (4 bytes)
    g1[1] = (int)((nelem & 0xFFFFu) << 16); // tensor_dim0[15:0] @ bits 63:48
    g1[2] = (int)((nelem >> 16) | (1u << 16)); // tensor_dim0[31:16] | tensor_dim1=1
    g1[3] = (int)(nelem << 16);             // tile_dim0 @ bits 127:112
    g1[4] = 1;                              // tile_dim1=1, tile_dim2=0
    g1[5] = (int)nelem;                     // tensor_dim0_stride[31:0]
    g1[6] = 0;                              // stride hi / dim1_stride
    g1[7] = 0;

    v4i gz = {0, 0, 0, 0};                  // groups 2/3 unused (<=2D tensor)
#if defined(TDM_6ARG)
    v8i gz8 = {0, 0, 0, 0, 0, 0, 0, 0};
    __builtin_amdgcn_tensor_load_to_lds(g0, g1, gz, gz, gz8, 0);
#else
    __builtin_amdgcn_tensor_load_to_lds(g0, g1, gz, gz, 0);
#endif
    __builtin_amdgcn_s_wait_tensorcnt((short)0);
  }
#else
  // Fallback: cooperative 16B global->LDS copy.
  {
    const v4f* src = (const v4f*)(x + (size_t)gbase * HH);
    v4f* dst = (v4f*)sm;
    for (int it = threadIdx.x; it < LDS_ROWS * F4_PER_ROW; it += THREADS)
      dst[it] = src[it];
  }
#endif
  __syncthreads();

  const v4f* smv = (const v4f*)sm;
  v4f* outv = (v4f*)out;

  for (int it = threadIdx.x; it < CHUNK * F4_PER_ROW; it += THREADS) {
    const int k = it >> 7;                  // row within chunk [0,16)
    const int f4 = it & (F4_PER_ROW - 1);   // float4 index within row
    const int p = r0 + k;                   // global flattened row
    const int lrow = p - gbase;             // LDS row of current row
    const int s = p & (SS - 1);             // sequence position within batch

    v4f cur = smv[lrow * F4_PER_ROW + f4];
    const bool sel = (bm[p >> 5] >> (p & 31u)) & 1u;

    v4f o = cur;
    if (sel) {
      // _shift pads with zeros at the start of each sequence, and those zeros
      // participate in the min — replicate exactly.
      v4f p1 = {0.f, 0.f, 0.f, 0.f};
      v4f p2 = {0.f, 0.f, 0.f, 0.f};
      if (s >= 1) p1 = smv[(lrow - 1) * F4_PER_ROW + f4];
      if (s >= 2) p2 = smv[(lrow - 2) * F4_PER_ROW + f4];
      o.x = fminf(fminf(cur.x, p1.x), p2.x);
      o.y = fminf(fminf(cur.y, p1.y), p2.y);
      o.z = fminf(fminf(cur.z, p1.z), p2.z);
      o.w = fminf(fminf(cur.w, p1.w), p2.w);
    }

    v4f r;
    {
      float g;
      g = sigmoid_f(cur.x); r.x = g * o.x + (1.0f - g) * cur.x;
      g = sigmoid_f(cur.y); r.y = g * o.y + (1.0f - g) * cur.y;
      g = sigmoid_f(cur.z); r.z = g * o.z + (1.0f - g) * cur.z;
      g = sigmoid_f(cur.w); r.w = g * o.w + (1.0f - g) * cur.w;
    }
    // Output is write-once / never re-read: non-temporal store (TH=NT) keeps
    // the 128 MB result from rinsing x's lines out of the 192 MB L2.
    __builtin_nontemporal_store(r, &outv[(size_t)p * F4_PER_ROW + f4]);
  }
}

// ---------------------------------------------------------------- bitmap ----
__global__ void zero_bitmap_kernel(unsigned* bm, int nwords) {
  int i = blockIdx.x * blockDim.x + threadIdx.x;
  if (i < nwords) bm[i] = 0u;
}

__global__ void set_bitmap_kernel(const int* __restrict__ idx, int n,
                                  unsigned* __restrict__ bm) {
  int i = blockIdx.x * blockDim.x + threadIdx.x;
  if (i < n) {
    unsigned p = (unsigned)idx[i];
    atomicOr(&bm[p >> 5], 1u << (p & 31u));
  }
}

// ---------------------------------------------------------------- launcher --
extern "C" void kernel_launch(void* const* d_in, const int* in_sizes, int n_in,
                              void* d_out, int out_size, void* d_ws,
                              size_t ws_size, hipStream_t stream) {
  const float* x = (const float*)d_in[0];
  const int* idx = (const int*)d_in[1];
  const int n_idx = in_sizes[1];

  unsigned* bm = (unsigned*)d_ws;        // 2048 words = 8 KB of scratch
  const int nwords = ROWS_TOTAL / 32;

  zero_bitmap_kernel<<<(nwords + 255) / 256, 256, 0, stream>>>(bm, nwords);
  set_bitmap_kernel<<<(n_idx + 255) / 256, 256, 0, stream>>>(idx, n_idx, bm);
  ngram_min_fused<<<ROWS_TOTAL / CHUNK, THREADS, 0, stream>>>(
      x, bm, (float*)d_out);
}